// MACE_layer_1297080123516
// MI455X (gfx1250) — compile-verified
//
#include <hip/hip_runtime.h>
#include <hip/hip_bf16.h>
#include <math.h>

#define NN 10000
#define EE 160000
#define CC 128
#define MLPDIM 64
#define FFDIM 128

typedef float v2f __attribute__((ext_vector_type(2)));
typedef float v8f __attribute__((ext_vector_type(8)));

static __device__ __forceinline__ v8f wmma4(v2f a, v2f b, v8f c) {
  // V_WMMA_F32_16X16X4_F32: D = A(16x4) * B(4x16) + C  (fp32 matrix path)
  return __builtin_amdgcn_wmma_f32_16x16x4_f32(false, a, false, b, (short)0, c,
                                               false, false);
}

// ---------------------------------------------------------------------------
__global__ void zero_kernel(float* __restrict__ p, int count) {
  int i = blockIdx.x * blockDim.x + threadIdx.x;
  int stride = gridDim.x * blockDim.x;
  for (; i < count; i += stride) p[i] = 0.f;
}

// ---------------------------------------------------------------------------
// [rows x 128] @ [128 x 128] GEMM. Each block: 80 rows (5 row-tiles/wave) x
// 128 cols (8 waves). A block staged in LDS once; B fragments amortized over
// 5 WMMA each. A element (n,k) at A[n*aRow + k*ACOL + aOff],
// Out element (n,c) at Out[n*oRow + c*oCol + oOff].
template <int ACOL>
__global__ __launch_bounds__(256) void gemm128T(
    const float* __restrict__ A, int aRow, int aOff,
    const float* __restrict__ W, float* __restrict__ Out, int oRow, int oCol,
    int oOff) {
  __shared__ float sA[80 * 132];
  const int tid = threadIdx.x;
  const int row0 = blockIdx.x * 80;
  for (int i = tid; i < 80 * 128; i += 256) {
    const int r = i >> 7, k = i & 127;
    sA[r * 132 + k] = A[(size_t)(row0 + r) * aRow + k * ACOL + aOff];
  }
  __syncthreads();
  const int lane = tid & 31;
  const int hl = lane >> 4;
  const int idx = lane & 15;
  const int cw = (tid >> 5) * 16;
  v8f acc[5] = {};
#pragma unroll 2
  for (int k = 0; k < 128; k += 4) {
    const int kk = k + 2 * hl;
    v2f b;
    b.x = W[kk * 128 + cw + idx];
    b.y = W[(kk + 1) * 128 + cw + idx];
#pragma unroll
    for (int t = 0; t < 5; ++t) {
      const v2f a = *(const v2f*)&sA[(t * 16 + idx) * 132 + kk];
      acc[t] = wmma4(a, b, acc[t]);
    }
  }
#pragma unroll
  for (int t = 0; t < 5; ++t)
#pragma unroll
    for (int r = 0; r < 8; ++r)
      Out[(size_t)(row0 + t * 16 + r + 8 * hl) * oRow + (cw + idx) * oCol +
          oOff] = acc[t][r];
}

// ---------------------------------------------------------------------------
// Fused per-edge pipeline (32 edges/block): async-DMA staging + sph harmonics
// + radial MLP (2 layers, WMMA) + messages + atomic scatter into agg.
__global__ __launch_bounds__(256) void edge_kernel(
    const float* __restrict__ vectors, const float* __restrict__ lengths,
    const float* __restrict__ edge_feats, const int* __restrict__ snd,
    const int* __restrict__ rcv, const float* __restrict__ Wr1,
    const float* __restrict__ br1, const float* __restrict__ Wr2,
    const float* __restrict__ h, float* __restrict__ agg) {
  __shared__ float sRin[32 * 136];  // cols 0..127 = edge_feats, col 128 = length
  __shared__ float sZ[32 * 68];     // post-SiLU layer-1 activations
  __shared__ float sY[32 * 8];      // Y1[0..2], Y2[0..4]
  __shared__ int sSnd[32], sRcv[32];

  const int tid = threadIdx.x;
  const int e0 = blockIdx.x * 32;

  // Stage the 32x128 edge_feats tile via CDNA5 async load-to-LDS (ASYNCcnt,
  // bypasses VGPRs). 4096 floats = 1024 x b128; 4 per thread.
#pragma unroll
  for (int i = 0; i < 4; ++i) {
    const int f = (i * 256 + tid) * 4;
    const int row = f >> 7, col = f & 127;
    const unsigned lds_off =
        (unsigned)(uintptr_t)(const void*)&sRin[row * 136 + col];
    const unsigned long long gaddr =
        (unsigned long long)(uintptr_t)(edge_feats +
                                        (size_t)(e0 + row) * 128 + col);
    asm volatile("global_load_async_to_lds_b128 %0, %1, off" ::"v"(lds_off),
                 "v"(gaddr)
                 : "memory");
  }
  if (tid < 32) {
    const int e = e0 + tid;
    sRin[tid * 136 + 128] = lengths[e];
    sSnd[tid] = snd[e];
    sRcv[tid] = rcv[e];
    float x = vectors[e * 3 + 0], y = vectors[e * 3 + 1], z = vectors[e * 3 + 2];
    const float inv = 1.f / (sqrtf(x * x + y * y + z * z) + 1e-9f);
    x *= inv; y *= inv; z *= inv;
    const float s3 = 1.7320508075688772f;    // sqrt(3)
    const float s5h = 1.1180339887498949f;   // sqrt(5)/2
    const float s15 = 3.872983346207417f;    // sqrt(15)
    const float s15h = 1.9364916731037085f;  // sqrt(15)/2
    float* Yp = sY + tid * 8;
    Yp[0] = s3 * x; Yp[1] = s3 * y; Yp[2] = s3 * z;
    Yp[3] = s15 * x * y;
    Yp[4] = s15 * y * z;
    Yp[5] = s5h * (3.f * z * z - 1.f);
    Yp[6] = s15 * x * z;
    Yp[7] = s15h * (x * x - y * y);
  }
  asm volatile("s_wait_asynccnt 0x0" ::: "memory");
  __syncthreads();

  const int lane = tid & 31;
  const int hl = lane >> 4;
  const int idx = lane & 15;
  const int wave = tid >> 5;

  // ---- radial MLP layer 1: [32,129] @ [129,64] + bias, SiLU ----
  // 8 waves = 2 row-tiles x 4 col-tiles.
  {
    const int rt = wave >> 2;          // row tile 0..1
    const int cw4 = (wave & 3) * 16;   // col tile base
    v8f acc = {};
#pragma unroll 4
    for (int k = 0; k < 128; k += 4) {
      const int kk = k + 2 * hl;
      const v2f a = *(const v2f*)&sRin[(rt * 16 + idx) * 136 + kk];
      v2f b;
      b.x = Wr1[kk * 64 + cw4 + idx];
      b.y = Wr1[(kk + 1) * 64 + cw4 + idx];
      acc = wmma4(a, b, acc);
    }
    const float wl = Wr1[128 * 64 + cw4 + idx];  // 129th input row (length)
    const float bias = br1[cw4 + idx];
#pragma unroll
    for (int r = 0; r < 8; ++r) {
      const int m = rt * 16 + r + 8 * hl;
      float x = acc[r] + sRin[m * 136 + 128] * wl + bias;
      x = x / (1.f + __expf(-x));  // SiLU
      sZ[m * 68 + cw4 + idx] = x;
    }
  }
  __syncthreads();

  // ---- radial MLP layer 2: [32,64] @ [64,3*128] ----
  // Each wave: one 16-channel tile, 2 edge-tiles x 3 l-paths = 6 accumulators
  // sharing each B fragment (6 WMMA per 6 global B floats + 4 LDS A floats).
  const int cw = wave * 16;
  v8f acc[2][3] = {};
#pragma unroll 4
  for (int k = 0; k < 64; k += 4) {
    const int kk = k + 2 * hl;
    const v2f a0 = *(const v2f*)&sZ[idx * 68 + kk];
    const v2f a1 = *(const v2f*)&sZ[(16 + idx) * 68 + kk];
    const float* w0p = Wr2 + kk * 384 + cw + idx;
    const float* w1p = Wr2 + (kk + 1) * 384 + cw + idx;
    v2f b0, b1v, b2;
    b0.x = w0p[0];    b0.y = w1p[0];
    b1v.x = w0p[128]; b1v.y = w1p[128];
    b2.x = w0p[256];  b2.y = w1p[256];
    acc[0][0] = wmma4(a0, b0, acc[0][0]);
    acc[1][0] = wmma4(a1, b0, acc[1][0]);
    acc[0][1] = wmma4(a0, b1v, acc[0][1]);
    acc[1][1] = wmma4(a1, b1v, acc[1][1]);
    acc[0][2] = wmma4(a0, b2, acc[0][2]);
    acc[1][2] = wmma4(a1, b2, acc[1][2]);
  }

  // ---- messages + scatter (pre-scaled by 1/AVG = 1/16) ----
  const int c = cw + idx;
#pragma unroll
  for (int t = 0; t < 2; ++t) {
#pragma unroll
    for (int r = 0; r < 8; ++r) {
      const int el = t * 16 + r + 8 * hl;
      const float hs = h[(size_t)sSnd[el] * 128 + c] * 0.0625f;
      const int b = sRcv[el];
      const float* Yp = sY + el * 8;
      unsafeAtomicAdd(&agg[(size_t)b * 128 + c], hs * acc[t][0][r]);
      const float p1 = hs * acc[t][1][r];
      float* a1 = agg + (size_t)NN * CC + ((size_t)b * 128 + c) * 3;
      unsafeAtomicAdd(a1 + 0, p1 * Yp[0]);
      unsafeAtomicAdd(a1 + 1, p1 * Yp[1]);
      unsafeAtomicAdd(a1 + 2, p1 * Yp[2]);
      const float p2 = hs * acc[t][2][r];
      float* a2 = agg + (size_t)4 * NN * CC + ((size_t)b * 128 + c) * 5;
      unsafeAtomicAdd(a2 + 0, p2 * Yp[3]);
      unsafeAtomicAdd(a2 + 1, p2 * Yp[4]);
      unsafeAtomicAdd(a2 + 2, p2 * Yp[5]);
      unsafeAtomicAdd(a2 + 3, p2 * Yp[6]);
      unsafeAtomicAdd(a2 + 4, p2 * Yp[7]);
    }
  }
}

// ---------------------------------------------------------------------------
// Symmetric contraction (correlation=3) per (node, channel). Pure VALU.
__global__ __launch_bounds__(256) void symcon(const float* __restrict__ svt,
                                              const float* __restrict__ w0,
                                              const float* __restrict__ w1,
                                              float* __restrict__ B) {
  const int gid = blockIdx.x * 256 + threadIdx.x;  // < N*C
  const int c = gid & 127;
  const float s = svt[gid];
  const float* vp = svt + (size_t)NN * CC + (size_t)gid * 3;
  const float* tp = svt + (size_t)4 * NN * CC + (size_t)gid * 5;
  const float vx = vp[0], vy = vp[1], vz = vp[2];
  const float a = tp[0], b = tp[1], c2 = tp[2], d = tp[3], e = tp[4];
  const float c3 = c2 * 0.5773502691896258f;  // c / sqrt(3)
  const float Tvx = (e - c3) * vx + a * vy + d * vz;
  const float Tvy = a * vx + (-e - c3) * vy + b * vz;
  const float Tvz = d * vx + b * vy + 2.f * c3 * vz;
  const float v2 = vx * vx + vy * vy + vz * vz;
  const float t2 = a * a + b * b + c2 * c2 + d * d + e * e;
  const float vTv = vx * Tvx + vy * Tvy + vz * Tvz;
  const float s2 = s * s;
  const float B0 = w0[0 * CC + c] * s + w0[1 * CC + c] * s2 +
                   w0[2 * CC + c] * v2 + w0[3 * CC + c] * t2 +
                   w0[4 * CC + c] * s2 * s + w0[5 * CC + c] * s * v2 +
                   w0[6 * CC + c] * s * t2 + w0[7 * CC + c] * vTv;
  const float k0 = w1[0 * CC + c], k1 = w1[1 * CC + c], k2 = w1[2 * CC + c];
  const float k3 = w1[3 * CC + c], k4 = w1[4 * CC + c], k5 = w1[5 * CC + c];
  const float cv = k0 + k1 * s + k3 * s2 + k4 * v2;  // coefficient on v
  const float ct = k2 + k5 * s;                      // coefficient on Tv
  B[gid] = B0;
  float* B1 = B + (size_t)NN * CC + (size_t)gid * 3;
  B1[0] = cv * vx + ct * Tvx;
  B1[1] = cv * vy + ct * Tvy;
  B1[2] = cv * vz + ct * Tvz;
}

// ---------------------------------------------------------------------------
// z = silu(h0 @ W1 + b1) : [N,128] @ [128,64]. 80 rows/block, 4 waves.
__global__ __launch_bounds__(128) void readout1(const float* __restrict__ out2,
                                                const float* __restrict__ W1,
                                                const float* __restrict__ b1,
                                                float* __restrict__ zbuf) {
  __shared__ float sA[80 * 132];
  const int tid = threadIdx.x;
  const int row0 = blockIdx.x * 80;
  for (int i = tid; i < 80 * 128; i += 128) {
    const int r = i >> 7, k = i & 127;
    sA[r * 132 + k] = out2[(size_t)(row0 + r) * 512 + k];
  }
  __syncthreads();
  const int lane = tid & 31;
  const int hl = lane >> 4;
  const int idx = lane & 15;
  const int cw = (tid >> 5) * 16;  // 0..48
  v8f acc[5] = {};
#pragma unroll 2
  for (int k = 0; k < 128; k += 4) {
    const int kk = k + 2 * hl;
    v2f b;
    b.x = W1[kk * 64 + cw + idx];
    b.y = W1[(kk + 1) * 64 + cw + idx];
#pragma unroll
    for (int t = 0; t < 5; ++t) {
      const v2f a = *(const v2f*)&sA[(t * 16 + idx) * 132 + kk];
      acc[t] = wmma4(a, b, acc[t]);
    }
  }
  const float bias = b1[cw + idx];
#pragma unroll
  for (int t = 0; t < 5; ++t)
#pragma unroll
    for (int r = 0; r < 8; ++r) {
      float x = acc[t][r] + bias;
      x = x / (1.f + __expf(-x));
      zbuf[(size_t)(row0 + t * 16 + r + 8 * hl) * 64 + cw + idx] = x;
    }
}

// scal = z @ W2 ; gate = z @ W3 ; vec = (h1 . wv) * gate. 80 rows/block.
__global__ __launch_bounds__(256) void readout2(
    const float* __restrict__ zbuf, const float* __restrict__ W2,
    const float* __restrict__ W3, const float* __restrict__ wv,
    const float* __restrict__ out2, float* __restrict__ out0,
    float* __restrict__ out1) {
  __shared__ float sZ[80 * 68];
  const int tid = threadIdx.x;
  const int row0 = blockIdx.x * 80;
  for (int i = tid; i < 80 * 64; i += 256) {
    const int r = i >> 6, k = i & 63;
    sZ[r * 68 + k] = zbuf[(size_t)(row0 + r) * 64 + k];
  }
  __syncthreads();
  const int lane = tid & 31;
  const int hl = lane >> 4;
  const int idx = lane & 15;
  const int cw = (tid >> 5) * 16;
  v8f acc[5] = {};
#pragma unroll 2
  for (int k = 0; k < 64; k += 4) {
    const int kk = k + 2 * hl;
    v2f b;
    b.x = W2[kk * 128 + cw + idx];
    b.y = W2[(kk + 1) * 128 + cw + idx];
#pragma unroll
    for (int t = 0; t < 5; ++t) {
      const v2f a = *(const v2f*)&sZ[(t * 16 + idx) * 68 + kk];
      acc[t] = wmma4(a, b, acc[t]);
    }
  }
#pragma unroll
  for (int t = 0; t < 5; ++t)
#pragma unroll
    for (int r = 0; r < 8; ++r)
      out0[(size_t)(row0 + t * 16 + r + 8 * hl) * 128 + cw + idx] = acc[t][r];

  if (tid < 80) {
    const int m = row0 + tid;
    float g = 0.f;
    const float* zr = sZ + tid * 68;
    for (int j = 0; j < 64; ++j) g += zr[j] * W3[j];
    const float* h1 = out2 + (size_t)m * 512 + 128;
    float vx = 0.f, vy = 0.f, vz = 0.f;
    for (int c = 0; c < 128; ++c) {
      const float wc = wv[c];
      vx += h1[c * 3 + 0] * wc;
      vy += h1[c * 3 + 1] * wc;
      vz += h1[c * 3 + 2] * wc;
    }
    out1[m * 3 + 0] = vx * g;
    out1[m * 3 + 1] = vy * g;
    out1[m * 3 + 2] = vz * g;
  }
}

// ---------------------------------------------------------------------------
extern "C" void kernel_launch(void* const* d_in, const int* in_sizes, int n_in,
                              void* d_out, int out_size, void* d_ws,
                              size_t ws_size, hipStream_t stream) {
  (void)in_sizes; (void)n_in; (void)out_size; (void)ws_size;
  const float* vectors = (const float*)d_in[0];
  const float* lengths = (const float*)d_in[1];
  const float* node_feats = (const float*)d_in[2];
  const float* edge_feats = (const float*)d_in[3];
  const int* edge_index = (const int*)d_in[4];
  const float* W_up = (const float*)d_in[5];
  const float* Wr1 = (const float*)d_in[6];
  const float* br1 = (const float*)d_in[7];
  const float* Wr2 = (const float*)d_in[8];
  const float* Wl0 = (const float*)d_in[9];
  const float* Wl1 = (const float*)d_in[10];
  const float* Wl2 = (const float*)d_in[11];
  const float* w0 = (const float*)d_in[12];
  const float* w1 = (const float*)d_in[13];
  const float* P0 = (const float*)d_in[14];
  const float* P1 = (const float*)d_in[15];
  const float* W1 = (const float*)d_in[16];
  const float* b1 = (const float*)d_in[17];
  const float* W2 = (const float*)d_in[18];
  const float* W3 = (const float*)d_in[19];
  const float* wv = (const float*)d_in[20];

  float* ws = (float*)d_ws;
  float* h = ws;                            // N*C
  float* agg = h + (size_t)NN * CC;         // 9*N*C  (agg0|agg1|agg2)
  float* svt = agg + (size_t)9 * NN * CC;   // 9*N*C  (s|v|t)
  float* Bb = svt + (size_t)9 * NN * CC;    // 4*N*C  (B0|B1)
  float* zb = Bb + (size_t)4 * NN * CC;     // N*64

  float* out0 = (float*)d_out;              // [N,128] scalars
  float* out1 = out0 + (size_t)NN * FFDIM;  // [N,3]   vector
  float* out2 = out1 + (size_t)NN * 3;      // [N,512] node_feats_out

  const int* snd = edge_index;
  const int* rcv = edge_index + EE;

  zero_kernel<<<2048, 256, 0, stream>>>(agg, 9 * NN * CC);

  // h = node_feats @ W_up
  gemm128T<1><<<NN / 80, 256, 0, stream>>>(node_feats, 128, 0, W_up, h, 128, 1,
                                           0);

  // fused edge pipeline + scatter (32 edges per block)
  edge_kernel<<<EE / 32, 256, 0, stream>>>(vectors, lengths, edge_feats, snd,
                                           rcv, Wr1, br1, Wr2, h, agg);

  // s = agg0@Wl0 ; v = agg1@Wl1 (x3) ; t = agg2@Wl2 (x5)
  gemm128T<1><<<NN / 80, 256, 0, stream>>>(agg, 128, 0, Wl0, svt, 128, 1, 0);
  for (int x = 0; x < 3; ++x)
    gemm128T<3><<<NN / 80, 256, 0, stream>>>(agg + (size_t)NN * CC, 384, x, Wl1,
                                             svt + (size_t)NN * CC, 384, 3, x);
  for (int x = 0; x < 5; ++x)
    gemm128T<5><<<NN / 80, 256, 0, stream>>>(agg + (size_t)4 * NN * CC, 640, x,
                                             Wl2, svt + (size_t)4 * NN * CC,
                                             640, 5, x);

  symcon<<<NN * CC / 256, 256, 0, stream>>>(svt, w0, w1, Bb);

  // h0 = B0@P0 ; h1 = B1@P1 (x3) -> node_feats_out region of d_out
  gemm128T<1><<<NN / 80, 256, 0, stream>>>(Bb, 128, 0, P0, out2, 512, 1, 0);
  for (int x = 0; x < 3; ++x)
    gemm128T<3><<<NN / 80, 256, 0, stream>>>(Bb + (size_t)NN * CC, 384, x, P1,
                                             out2, 512, 3, 128 + x);

  readout1<<<NN / 80, 128, 0, stream>>>(out2, W1, b1, zb);
  readout2<<<NN / 80, 256, 0, stream>>>(zb, W2, W3, wv, out2, out0, out1);
}